// PPGAT_72009421684886
// MI455X (gfx1250) — compile-verified
//
#include <hip/hip_runtime.h>

// ---------------------------------------------------------------------------
// PPGAT on gfx1250: GATv2 x4 + pooling + MLP.
// Dense projections use V_WMMA_F32_16X16X4_F32 (fp32 WMMA, full precision).
// Edge softmax: atomic-max (monotone uint map) -> exp/atomic-add den -> agg.
// ---------------------------------------------------------------------------

typedef __attribute__((ext_vector_type(2))) float v2f;
typedef __attribute__((ext_vector_type(8))) float v8f;

#define DEVFN static __device__ __forceinline__

DEVFN unsigned fmono(float f) {
  unsigned u = __float_as_uint(f);
  return (u & 0x80000000u) ? ~u : (u | 0x80000000u);
}
DEVFN float fdemono(unsigned m) {
  unsigned u = (m & 0x80000000u) ? (m ^ 0x80000000u) : ~m;
  return __uint_as_float(u);
}

// ---------------------------------------------------------------------------
// GEMM: Y[M,N] = X[M,K] @ W[K,N] (+bias) (+relu).
// One wave computes a 16x64 strip: 4 f32-WMMA accumulators sharing one A frag.
// REQUIRES: M % 16 == 0, K % 4 == 0, N % 64 == 0 (true for every call here).
// A layout: lane&15 = row M, lane>>4 selects K-pair {0,1}/{2,3} (VGPR0/1).
// B layout: lane&15 = col N, lane>>4 selects K-pair (mirrors A).
// C/D layout: lane&15 = col, VGPR r -> row (lane>>4)*8 + r.
// ---------------------------------------------------------------------------
__global__ void k_gemm_wmma(const float* __restrict__ X, const float* __restrict__ W,
                            const float* __restrict__ bias, float* __restrict__ Y,
                            int M, int K, int N, int act) {
  const int lane = threadIdx.x & 31;
  const int half = lane >> 4;
  const int l15  = lane & 15;
  int wave = (int)((blockIdx.x * (long)blockDim.x + threadIdx.x) >> 5);
  int nw   = (int)(((long)gridDim.x * blockDim.x) >> 5);
  const int Mt = M >> 4;
  const int Nt = N >> 6;
  const long total = (long)Mt * Nt;
  for (long t = wave; t < total; t += nw) {
    const int tm = (int)(t % Mt);
    const int tn = (int)(t / Mt);
    const int colbase = tn * 64 + l15;
    const float* ap = X + (long)(tm * 16 + l15) * K + half * 2;
    const float* bp = W + (long)(half * 2) * N + colbase;
    v8f acc0 = {0.f, 0.f, 0.f, 0.f, 0.f, 0.f, 0.f, 0.f};
    v8f acc1 = acc0, acc2 = acc0, acc3 = acc0;
    for (int k0 = 0; k0 < K; k0 += 4) {
      v2f a = *(const v2f*)ap;  // two consecutive K elems -> global_load_b64
      ap += 4;
      v2f b0, b1, b2, b3;
      b0[0] = bp[0];  b0[1] = bp[N];
      b1[0] = bp[16]; b1[1] = bp[N + 16];
      b2[0] = bp[32]; b2[1] = bp[N + 32];
      b3[0] = bp[48]; b3[1] = bp[N + 48];
      bp += (long)4 * N;
      acc0 = __builtin_amdgcn_wmma_f32_16x16x4_f32(false, a, false, b0, (short)0, acc0, false, false);
      acc1 = __builtin_amdgcn_wmma_f32_16x16x4_f32(false, a, false, b1, (short)0, acc1, false, false);
      acc2 = __builtin_amdgcn_wmma_f32_16x16x4_f32(false, a, false, b2, (short)0, acc2, false, false);
      acc3 = __builtin_amdgcn_wmma_f32_16x16x4_f32(false, a, false, b3, (short)0, acc3, false, false);
    }
    float bb0 = 0.f, bb1 = 0.f, bb2 = 0.f, bb3 = 0.f;
    if (bias) {
      bb0 = bias[colbase];      bb1 = bias[colbase + 16];
      bb2 = bias[colbase + 32]; bb3 = bias[colbase + 48];
    }
#pragma unroll
    for (int r = 0; r < 8; ++r) {
      const int orow = tm * 16 + half * 8 + r;
      float* yr = Y + (long)orow * N + colbase;
      float v0 = acc0[r] + bb0, v1 = acc1[r] + bb1;
      float v2 = acc2[r] + bb2, v3 = acc3[r] + bb3;
      if (act == 1) {
        v0 = fmaxf(v0, 0.f); v1 = fmaxf(v1, 0.f);
        v2 = fmaxf(v2, 0.f); v3 = fmaxf(v3, 0.f);
      }
      yr[0] = v0; yr[16] = v1; yr[32] = v2; yr[48] = v3;
    }
  }
}

// ---------------------------------------------------------------------------
__global__ void k_zero(float* __restrict__ p, long n) {
  long i = (long)blockIdx.x * blockDim.x + threadIdx.x;
  long st = (long)gridDim.x * blockDim.x;
  for (; i < n; i += st) p[i] = 0.f;
}

__global__ void k_bias_rows(float* __restrict__ out, const float* __restrict__ bias,
                            long n, int C) {
  long i = (long)blockIdx.x * blockDim.x + threadIdx.x;
  long st = (long)gridDim.x * blockDim.x;
  for (; i < n; i += st) out[i] = bias[i % C];
}

__global__ void k_elu(float* __restrict__ p, long n) {
  long i = (long)blockIdx.x * blockDim.x + threadIdx.x;
  long st = (long)gridDim.x * blockDim.x;
  for (; i < n; i += st) {
    float v = p[i];
    p[i] = v > 0.f ? v : expm1f(v);
  }
}

// p[i] /= max(cnt[i/C],1)
__global__ void k_div_rows(float* __restrict__ p, const float* __restrict__ cnt,
                           long n, int C) {
  long i = (long)blockIdx.x * blockDim.x + threadIdx.x;
  long st = (long)gridDim.x * blockDim.x;
  for (; i < n; i += st) p[i] /= fmaxf(cnt[i / C], 1.f);
}

// ---------------------------------------------------------------------------
// Self-loop edge-attr mean: lsum[d] += ea[e], deg[d] += 1. Wave per edge, 64 ch.
__global__ void k_loop_accum(const int* __restrict__ dst, const float* __restrict__ ea,
                             float* __restrict__ lsum, float* __restrict__ deg, int E) {
  const int lane = threadIdx.x & 31;
  int wave = (int)((blockIdx.x * (long)blockDim.x + threadIdx.x) >> 5);
  int nw   = (int)(((long)gridDim.x * blockDim.x) >> 5);
  for (int e = wave; e < E; e += nw) {
    int d = dst[e];
#pragma unroll
    for (int j = 0; j < 2; ++j) {
      int c = lane * 2 + j;
      atomicAdd(lsum + (long)d * 64 + c, ea[(long)e * 64 + c]);
    }
    if (lane == 0) atomicAdd(deg + d, 1.f);
  }
}

// ---------------------------------------------------------------------------
// GATv2 score pass (wave per edge incl. self-loops). m = xl[s]+xr[d](+ea@we);
// score[e,h] = sum_c lrelu(m)*att ; atomic-max into smax (monotone uint).
__global__ void k_gat_score(const int* __restrict__ src, const int* __restrict__ dst,
                            int E, int Nn, int total, int H, int HC,
                            const float* __restrict__ xl, const float* __restrict__ xr,
                            const float* __restrict__ eacat, const float* __restrict__ we,
                            const float* __restrict__ att,
                            float* __restrict__ score, unsigned* __restrict__ smax) {
  const int lane = threadIdx.x & 31;
  int wave = (int)((blockIdx.x * (long)blockDim.x + threadIdx.x) >> 5);
  int nw   = (int)(((long)gridDim.x * blockDim.x) >> 5);
  const int P  = HC >> 5;   // channels per lane
  const int Lh = 32 / H;    // lanes per head (channels of one head are contiguous)
  for (int e = wave; e < total; e += nw) {
    int s, d;
    if (e < E) { s = src[e]; d = dst[e]; } else { s = e - E; d = s; }
    if (d >= Nn) continue;  // dropped dummy target (uniform per wave)
    float part = 0.f;
    for (int j = 0; j < P; ++j) {
      const int c = lane * P + j;
      float m = xl[(long)s * HC + c] + xr[(long)d * HC + c];
      if (eacat) {
        const float* ar = eacat + (long)e * 64;
        const float* wc = we + c;
        float acc = 0.f;
#pragma unroll 8
        for (int k = 0; k < 64; ++k) acc += ar[k] * wc[(long)k * HC];
        m += acc;
      }
      float lr = m > 0.f ? m : 0.2f * m;
      part += lr * att[c];  // att flat [H*Ch] == [HC]
    }
    for (int off = 1; off < Lh; off <<= 1) part += __shfl_xor(part, off, 32);
    if ((lane & (Lh - 1)) == 0) {
      const int h = lane / Lh;
      score[(long)e * H + h] = part;
      atomicMax(smax + (long)d * H + h, fmono(part));
    }
  }
}

__global__ void k_smax_decode(unsigned* __restrict__ p, long n) {
  long i = (long)blockIdx.x * blockDim.x + threadIdx.x;
  long st = (long)gridDim.x * blockDim.x;
  for (; i < n; i += st) {
    float f = fdemono(p[i]);
    ((float*)p)[i] = f;
  }
}

// ex = exp(score - smax[d]); score <- ex ; den[d] += ex.
__global__ void k_gat_den(const int* __restrict__ dst, int E, int Nn, int total, int H,
                          const float* __restrict__ smaxf, float* __restrict__ score,
                          float* __restrict__ den) {
  int i = blockIdx.x * blockDim.x + threadIdx.x;
  int st = gridDim.x * blockDim.x;
  for (int e = i; e < total; e += st) {
    int d = e < E ? dst[e] : e - E;
    if (d >= Nn) continue;
    for (int h = 0; h < H; ++h) {
      float ex = expf(score[(long)e * H + h] - smaxf[(long)d * H + h]);
      score[(long)e * H + h] = ex;
      atomicAdd(den + (long)d * H + h, ex);
    }
  }
}

// out[d] += (ex/den[d]) * xl[s]  (out pre-seeded with bias rows)
__global__ void k_gat_agg(const int* __restrict__ src, const int* __restrict__ dst,
                          int E, int Nn, int total, int H, int HC,
                          const float* __restrict__ xl, const float* __restrict__ ex,
                          const float* __restrict__ den, float* __restrict__ out) {
  const int lane = threadIdx.x & 31;
  int wave = (int)((blockIdx.x * (long)blockDim.x + threadIdx.x) >> 5);
  int nw   = (int)(((long)gridDim.x * blockDim.x) >> 5);
  const int P  = HC >> 5;
  const int Ch = HC / H;
  for (int e = wave; e < total; e += nw) {
    int s, d;
    if (e < E) { s = src[e]; d = dst[e]; } else { s = e - E; d = s; }
    if (d >= Nn) continue;
    for (int j = 0; j < P; ++j) {
      const int c = lane * P + j;
      const int h = c / Ch;
      float alpha = ex[(long)e * H + h] / den[(long)d * H + h];
      atomicAdd(out + (long)d * HC + c, alpha * xl[(long)s * HC + c]);
    }
  }
}

// ---------------------------------------------------------------------------
// Pharmacophore pooling: gx[g] += h[n]; gcnt[g]++; rbsum[g] += batch[n].
__global__ void k_pool_accum(const int* __restrict__ pharma, const int* __restrict__ batch,
                             const float* __restrict__ h, float* __restrict__ gx,
                             float* __restrict__ gcnt, float* __restrict__ rbsum, int Nn) {
  const int lane = threadIdx.x & 31;
  int wave = (int)((blockIdx.x * (long)blockDim.x + threadIdx.x) >> 5);
  int nw   = (int)(((long)gridDim.x * blockDim.x) >> 5);
  for (int n = wave; n < Nn; n += nw) {
    int g = pharma[n];
#pragma unroll
    for (int j = 0; j < 2; ++j) {
      int c = lane * 2 + j;
      atomicAdd(gx + (long)g * 64 + c, h[(long)n * 64 + c]);
    }
    if (lane == 0) {
      atomicAdd(gcnt + g, 1.f);
      atomicAdd(rbsum + g, (float)batch[n]);
    }
  }
}

__global__ void k_rbatch(const float* __restrict__ rbsum, const float* __restrict__ gcnt,
                         int* __restrict__ rbatch, int Gn) {
  int i = blockIdx.x * blockDim.x + threadIdx.x;
  int st = gridDim.x * blockDim.x;
  for (; i < Gn; i += st) rbatch[i] = (int)(rbsum[i] / fmaxf(gcnt[i], 1.f));
}

__global__ void k_edge_group(const int* __restrict__ src, const int* __restrict__ dst,
                             const int* __restrict__ pharma, int* __restrict__ sg,
                             int* __restrict__ dgm, int E, int Gn) {
  int i = blockIdx.x * blockDim.x + threadIdx.x;
  int st = gridDim.x * blockDim.x;
  for (; i < E; i += st) {
    int s = pharma[src[i]];
    int d = pharma[dst[i]];
    sg[i] = s;
    dgm[i] = (s == d) ? Gn : d;
  }
}

// Graph pooling: gsum[rbatch[g]] += h[g]; gn[rbatch[g]]++.
__global__ void k_gpool_accum(const int* __restrict__ rbatch, const float* __restrict__ h,
                              float* __restrict__ gsum, float* __restrict__ gn, int Gn) {
  const int lane = threadIdx.x & 31;
  int wave = (int)((blockIdx.x * (long)blockDim.x + threadIdx.x) >> 5);
  int nw   = (int)(((long)gridDim.x * blockDim.x) >> 5);
  for (int g = wave; g < Gn; g += nw) {
    int b = rbatch[g];
#pragma unroll
    for (int j = 0; j < 2; ++j) {
      int c = lane * 2 + j;
      atomicAdd(gsum + (long)b * 64 + c, h[(long)g * 64 + c]);
    }
    if (lane == 0) atomicAdd(gn + b, 1.f);
  }
}

// out[b] = z[b] . w2 + b2  (wave per graph)
__global__ void k_out(const float* __restrict__ z, const float* __restrict__ w2,
                      const float* __restrict__ b2, float* __restrict__ out, int B) {
  const int lane = threadIdx.x & 31;
  int wave = (int)((blockIdx.x * (long)blockDim.x + threadIdx.x) >> 5);
  int nw   = (int)(((long)gridDim.x * blockDim.x) >> 5);
  for (int b = wave; b < B; b += nw) {
    float part = z[(long)b * 64 + lane * 2] * w2[lane * 2] +
                 z[(long)b * 64 + lane * 2 + 1] * w2[lane * 2 + 1];
    for (int off = 1; off < 32; off <<= 1) part += __shfl_xor(part, off, 32);
    if (lane == 0) out[b] = part + b2[0];
  }
}

// ---------------------------------------------------------------------------
extern "C" void kernel_launch(void* const* d_in, const int* in_sizes, int n_in,
                              void* d_out, int out_size, void* d_ws, size_t ws_size,
                              hipStream_t stream) {
  (void)in_sizes; (void)n_in; (void)out_size; (void)ws_size;
  const int E = 400000, NN = 50000, G = 10000, NG = 256;

  const float* x         = (const float*)d_in[0];
  const int*   eidx      = (const int*)d_in[1];
  const int*   src       = eidx;
  const int*   dst       = eidx + E;
  const float* edge_attr = (const float*)d_in[2];
  const int*   batch     = (const int*)d_in[3];
  const int*   pharma    = (const int*)d_in[4];
  const float* w_edge = (const float*)d_in[6];
  const float* b_edge = (const float*)d_in[7];
  const float* g1_wl  = (const float*)d_in[8];
  const float* g1_wr  = (const float*)d_in[9];
  const float* g1_att = (const float*)d_in[10];
  const float* g1_we  = (const float*)d_in[11];
  const float* g1_b   = (const float*)d_in[12];
  const float* g2_wl  = (const float*)d_in[13];
  const float* g2_wr  = (const float*)d_in[14];
  const float* g2_att = (const float*)d_in[15];
  const float* g2_we  = (const float*)d_in[16];
  const float* g2_b   = (const float*)d_in[17];
  const float* g3_wl  = (const float*)d_in[18];
  const float* g3_wr  = (const float*)d_in[19];
  const float* g3_att = (const float*)d_in[20];
  const float* g3_b   = (const float*)d_in[21];
  const float* g4_wl  = (const float*)d_in[22];
  const float* g4_wr  = (const float*)d_in[23];
  const float* g4_att = (const float*)d_in[24];
  const float* g4_b   = (const float*)d_in[25];
  const float* w1     = (const float*)d_in[26];
  const float* b1     = (const float*)d_in[27];
  const float* w2     = (const float*)d_in[28];
  const float* b2     = (const float*)d_in[29];
  float* out = (float*)d_out;

  // --- workspace carve-out (~338 MB of f32) ---
  float* wsf = (float*)d_ws;
  size_t off = 0;
  auto A = [&](size_t n) {
    float* p = wsf + off;
    off += (n + 63) & ~(size_t)63;
    return p;
  };
  float* ea_cat = A((size_t)(E + NN) * 64);  // embedded edge attr + loop means
  float* xl1 = A((size_t)NN * 256);
  float* xr1 = A((size_t)NN * 256);
  float* h1  = A((size_t)NN * 256);
  float* xl2 = A((size_t)NN * 64);
  float* xr2 = A((size_t)NN * 64);
  float* h2  = A((size_t)NN * 64);
  float* score = A((size_t)(E + NN) * 4);
  float* smax  = A((size_t)NN * 4);  // uint during max, decoded to float in place
  float* den   = A((size_t)NN * 4);
  float* deg   = A(NN);
  int*   sg    = (int*)A(E);
  int*   dgm   = (int*)A(E);
  float* gx    = A((size_t)G * 64);
  float* gcnt  = A(G);
  float* rbsum = A(G);
  int*   rbatch = (int*)A(G);
  float* xl3 = A((size_t)G * 64);
  float* xr3 = A((size_t)G * 64);
  float* h3  = A((size_t)G * 64);
  float* xl4 = A((size_t)G * 64);
  float* xr4 = A((size_t)G * 64);
  float* h4  = A((size_t)G * 64);
  float* gsum = A((size_t)NG * 64);
  float* gn   = A(NG);
  float* z    = A((size_t)NG * 64);

  const int TPB = 256;
  auto blk_t = [](long n) { long b = (n + 255) / 256; if (b > (1L << 20)) b = 1L << 20; if (b < 1) b = 1; return (int)b; };
  auto blk_w = [](long w) { long b = (w + 7) / 8;     if (b > (1L << 20)) b = 1L << 20; if (b < 1) b = 1; return (int)b; };

  auto gemm = [&](const float* X, const float* W, const float* bias, float* Y,
                  int M, int K, int N, int act) {
    long tiles = (long)(M / 16) * (N / 64);  // M%16==0, N%64==0 for all calls
    k_gemm_wmma<<<blk_w(tiles), TPB, 0, stream>>>(X, W, bias, Y, M, K, N, act);
  };

  // One full GATv2 layer (score -> softmax -> aggregate -> elu).
  auto gat = [&](const int* s_, const int* d_, int Eedges, int Nn, int H, int HC,
                 const float* xl, const float* xr, const float* eac, const float* we,
                 const float* att, const float* bias, float* outbuf) {
    int total = Eedges + Nn;
    k_zero<<<blk_t((long)Nn * H), TPB, 0, stream>>>(smax, (long)Nn * H);
    k_zero<<<blk_t((long)Nn * H), TPB, 0, stream>>>(den, (long)Nn * H);
    k_gat_score<<<blk_w(total), TPB, 0, stream>>>(s_, d_, Eedges, Nn, total, H, HC,
                                                  xl, xr, eac, we, att, score,
                                                  (unsigned*)smax);
    k_smax_decode<<<blk_t((long)Nn * H), TPB, 0, stream>>>((unsigned*)smax, (long)Nn * H);
    k_gat_den<<<blk_t(total), TPB, 0, stream>>>(d_, Eedges, Nn, total, H, smax, score, den);
    k_bias_rows<<<blk_t((long)Nn * HC), TPB, 0, stream>>>(outbuf, bias, (long)Nn * HC, HC);
    k_gat_agg<<<blk_w(total), TPB, 0, stream>>>(s_, d_, Eedges, Nn, total, H, HC,
                                                xl, score, den, outbuf);
    k_elu<<<blk_t((long)Nn * HC), TPB, 0, stream>>>(outbuf, (long)Nn * HC);
  };

  // 1) edge embedding: ea = edge_attr @ w_edge + b_edge
  gemm(edge_attr, w_edge, b_edge, ea_cat, E, 16, 64, 0);

  // 2) self-loop attr = segment_mean(ea, dst) into ea_cat[E:]
  k_zero<<<blk_t((long)NN * 64), TPB, 0, stream>>>(ea_cat + (long)E * 64, (long)NN * 64);
  k_zero<<<blk_t(NN), TPB, 0, stream>>>(deg, NN);
  k_loop_accum<<<blk_w(E), TPB, 0, stream>>>(dst, ea_cat, ea_cat + (long)E * 64, deg, E);
  k_div_rows<<<blk_t((long)NN * 64), TPB, 0, stream>>>(ea_cat + (long)E * 64, deg,
                                                       (long)NN * 64, 64);

  // 3) layer 1 (H=4, C=64, edge features)
  gemm(x, g1_wl, nullptr, xl1, NN, 64, 256, 0);
  gemm(x, g1_wr, nullptr, xr1, NN, 64, 256, 0);
  gat(src, dst, E, NN, 4, 256, xl1, xr1, ea_cat, g1_we, g1_att, g1_b, h1);

  // 4) layer 2 (H=1, C=64, edge features)
  gemm(h1, g2_wl, nullptr, xl2, NN, 256, 64, 0);
  gemm(h1, g2_wr, nullptr, xr2, NN, 256, 64, 0);
  gat(src, dst, E, NN, 1, 64, xl2, xr2, ea_cat, g2_we, g2_att, g2_b, h2);

  // 5) pharmacophore pooling
  k_zero<<<blk_t((long)G * 64), TPB, 0, stream>>>(gx, (long)G * 64);
  k_zero<<<blk_t(G), TPB, 0, stream>>>(gcnt, G);
  k_zero<<<blk_t(G), TPB, 0, stream>>>(rbsum, G);
  k_pool_accum<<<blk_w(NN), TPB, 0, stream>>>(pharma, batch, h2, gx, gcnt, rbsum, NN);
  k_div_rows<<<blk_t((long)G * 64), TPB, 0, stream>>>(gx, gcnt, (long)G * 64, 64);
  k_rbatch<<<blk_t(G), TPB, 0, stream>>>(rbsum, gcnt, rbatch, G);
  k_edge_group<<<blk_t(E), TPB, 0, stream>>>(src, dst, pharma, sg, dgm, E, G);

  // 6) layer 3 (groups, H=1, no edge attr)
  gemm(gx, g3_wl, nullptr, xl3, G, 64, 64, 0);
  gemm(gx, g3_wr, nullptr, xr3, G, 64, 64, 0);
  gat(sg, dgm, E, G, 1, 64, xl3, xr3, nullptr, nullptr, g3_att, g3_b, h3);

  // 7) layer 4
  gemm(h3, g4_wl, nullptr, xl4, G, 64, 64, 0);
  gemm(h3, g4_wr, nullptr, xr4, G, 64, 64, 0);
  gat(sg, dgm, E, G, 1, 64, xl4, xr4, nullptr, nullptr, g4_att, g4_b, h4);

  // 8) global mean pool + MLP head
  k_zero<<<blk_t((long)NG * 64), TPB, 0, stream>>>(gsum, (long)NG * 64);
  k_zero<<<blk_t(NG), TPB, 0, stream>>>(gn, NG);
  k_gpool_accum<<<blk_w(G), TPB, 0, stream>>>(rbatch, h4, gsum, gn, G);
  k_div_rows<<<blk_t((long)NG * 64), TPB, 0, stream>>>(gsum, gn, (long)NG * 64, 64);
  gemm(gsum, w1, b1, z, NG, 64, 64, 1);  // relu
  k_out<<<blk_w(NG), TPB, 0, stream>>>(z, w2, b2, out, NG);
}